// ASIC_44186623541335
// MI455X (gfx1250) — compile-verified
//
#include <hip/hip_runtime.h>
#include <stdint.h>

#define GRID_N   16384
#define NMASK    (GRID_N - 1)
#define LAYERS   8
#define NPOS     32
#define MWIDTH   4096
#define TILE     64
#define HALO     16
#define WND      (TILE + 2 * HALO)       // 96 cells in LDS window
#define THREADS  256
#define CHUNK    ((WND * 32) / THREADS)  // 12 elements per thread per tile pass

// float-index offsets into dynamic LDS (all 128B aligned)
#define ST0_OFF  0                        // state ping  [WND][32]
#define ST1_OFF  (WND * 32)               // state pong  [WND][32]
#define RAW_OFF  (2 * WND * 32)           // raw tw stage [WND][32] (async target)
#define SIG_OFF  (3 * WND * 32)           // sigmoid(tw)  [WND][32]
#define SMEM_FLOATS (4 * WND * 32)        // 12288 floats = 48 KB

__device__ __forceinline__ void async_ld_b32(uint32_t lds_off, const float* base,
                                             uint32_t goff_bytes) {
    // GVS mode: mem_addr = SGPR64 + VGPR32; VDST VGPR carries the LDS byte address.
    asm volatile("global_load_async_to_lds_b32 %0, %1, %2"
                 :: "v"(lds_off), "v"(goff_bytes), "s"(base)
                 : "memory");
}
__device__ __forceinline__ void wait_async0() {
    asm volatile("s_wait_asynccnt 0" ::: "memory");
}

__global__ void __launch_bounds__(THREADS)
ASIC_toggle_grid_fused(const float* __restrict__ x,
                       const float* __restrict__ tg,
                       float* __restrict__ out) {
    extern __shared__ float smem[];
    const int tid  = threadIdx.x;
    const int lane = tid & 31;            // lane == batch index (wave32)
    const int wv   = tid >> 5;            // 8 waves per block
    const int g0   = blockIdx.x * TILE;   // tile start on the ring
    const int gs   = g0 - HALO + GRID_N;  // window start (pre-biased for wrap)

    // ---- embed: state[b][n] = x[b][n/4] if n%4==0 else 0 ----
    #pragma unroll
    for (int k = 0; k < CHUNK; ++k) {
        const int idx = tid + k * THREADS;
        const int j = idx >> 5, b = idx & 31;
        const int n = (gs + j) & NMASK;
        float v = 0.0f;
        if ((n & 3) == 0) v = x[b * MWIDTH + (n >> 2)];
        smem[ST0_OFF + idx] = v;
    }

    // ---- async-prefetch layer 0 toggle gates, transposed into raw[j][c] ----
    #pragma unroll
    for (int k = 0; k < CHUNK; ++k) {
        const int idx = tid + k * THREADS;
        const int c = idx & 31, j = idx >> 5;
        const int n = (gs + j) & NMASK;
        const uint32_t goff = (uint32_t)((unsigned)(c * GRID_N + n) * 4u);
        const uint32_t loff = (uint32_t)(uintptr_t)(&smem[RAW_OFF + idx]);
        async_ld_b32(loff, tg, goff);
    }

    const float* st_in  = &smem[ST0_OFF];
    float*       st_out = &smem[ST1_OFF];

    for (int l = 0; l < LAYERS; ++l) {
        wait_async0();              // this wave's async LDS writes have landed
        __syncthreads();            // publish raw tile to all waves

        // ---- sigmoid pass: sig = 1/(1+exp(-raw)) ----
        #pragma unroll
        for (int k = 0; k < CHUNK; ++k) {
            const int idx = tid + k * THREADS;
            const float v = smem[RAW_OFF + idx];
            smem[SIG_OFF + idx] = 1.0f / (1.0f + __expf(-v));
        }
        __syncthreads();            // raw fully consumed -> safe to refill

        // ---- kick async prefetch of next layer's tw while we compute ----
        if (l + 1 < LAYERS) {
            #pragma unroll
            for (int k = 0; k < CHUNK; ++k) {
                const int idx = tid + k * THREADS;
                const int c = idx & 31, j = idx >> 5;
                const int n = (gs + j) & NMASK;
                const uint32_t goff =
                    (uint32_t)(((unsigned)((l + 1) * NPOS + c) * GRID_N + (unsigned)n) * 4u);
                const uint32_t loff = (uint32_t)(uintptr_t)(&smem[RAW_OFF + idx]);
                async_ld_b32(loff, tg, goff);
            }
        }

        // ---- compute layer l with shrinking halo; lane == batch ----
        const int lo = 2 * (l + 1), hi = WND - 2 * (l + 1);
        for (int j = lo + wv; j < hi; j += 8) {     // wave-uniform j
            // 32 lane-uniform coefficients (LDS broadcast, 8x ds_load_b128)
            float t[32];
            const float4* cp = (const float4*)&smem[SIG_OFF + j * 32];
            #pragma unroll
            for (int q = 0; q < 8; ++q) {
                float4 v = cp[q];
                t[4*q+0] = v.x; t[4*q+1] = v.y; t[4*q+2] = v.z; t[4*q+3] = v.w;
            }
            // neighbor states for this lane's batch (conflict-free rows)
            const float s0 = st_in[(j - 2) * 32 + lane];
            const float s1 = st_in[(j - 1) * 32 + lane];
            const float s2 = st_in[(j    ) * 32 + lane];
            const float s3 = st_in[(j + 1) * 32 + lane];
            const float s4 = st_in[(j + 2) * 32 + lane];
            // multilinear fold: LSB of combo index == kernel tap i=4 (n+2) first
            #pragma unroll
            for (int k = 0; k < 16; ++k) t[k] = fmaf(s4, t[2*k+1] - t[2*k], t[2*k]);
            #pragma unroll
            for (int k = 0; k < 8;  ++k) t[k] = fmaf(s3, t[2*k+1] - t[2*k], t[2*k]);
            #pragma unroll
            for (int k = 0; k < 4;  ++k) t[k] = fmaf(s2, t[2*k+1] - t[2*k], t[2*k]);
            #pragma unroll
            for (int k = 0; k < 2;  ++k) t[k] = fmaf(s1, t[2*k+1] - t[2*k], t[2*k]);
            float r = fmaf(s0, t[1] - t[0], t[0]);
            r = fminf(fmaxf(r, 0.0f), 1.0f);
            st_out[j * 32 + lane] = r;
        }
        __syncthreads();
        // swap ping/pong
        const float* tmp_in = st_in;
        st_in  = st_out;
        st_out = (float*)tmp_in;
    }

    // After 8 layers st_in points at the final state (ST0).
    // out[b][m] = state[b][4m]; tile contributes m = g0/4 .. g0/4 + 15
    #pragma unroll
    for (int k = 0; k < (TILE / 4) * 32 / THREADS; ++k) {
        const int idx = tid + k * THREADS;
        const int q = idx >> 5, b = idx & 31;
        const int j = HALO + 4 * q;
        out[b * MWIDTH + (g0 >> 2) + q] = st_in[j * 32 + b];
    }
}

extern "C" void kernel_launch(void* const* d_in, const int* in_sizes, int n_in,
                              void* d_out, int out_size, void* d_ws, size_t ws_size,
                              hipStream_t stream) {
    const float* x  = (const float*)d_in[0];   // (32, 4096) f32
    const float* tg = (const float*)d_in[1];   // (8, 32, 16384) f32
    float* out = (float*)d_out;                // (32, 4096) f32
    dim3 grid(GRID_N / TILE);                  // 256 blocks
    dim3 block(THREADS);                       // 8 waves (wave32)
    ASIC_toggle_grid_fused<<<grid, block, SMEM_FLOATS * sizeof(float), stream>>>(x, tg, out);
}